// MultiHeadGAT_79491254714423
// MI455X (gfx1250) — compile-verified
//
#include <hip/hip_runtime.h>
#include <hip/hip_bf16.h>

typedef float v2f __attribute__((ext_vector_type(2)));
typedef float v8f __attribute__((ext_vector_type(8)));
typedef int   v4i __attribute__((vector_size(16)));   // matches builtin param type

#define Bsz   8
#define Cdim  1024
#define INF   256
#define NH    8
#define OF    128
#define NF    1024      // NH*OF
#define TEMP_INV (1.0f/1.5f)
#define LN_EPS 1e-5f

// ---- CDNA5 async global->LDS copy (guarded; falls back to reg copy) ----
__device__ __forceinline__ void g2l_b128(const float* g, float* l) {
#if defined(__gfx1250__) && __has_builtin(__builtin_amdgcn_global_load_async_to_lds_b128)
    __builtin_amdgcn_global_load_async_to_lds_b128(
        (__attribute__((address_space(1))) v4i*)g,
        (__attribute__((address_space(3))) v4i*)l, 0, 0);
#else
    *(float4*)l = *(const float4*)g;
#endif
}
__device__ __forceinline__ void async_waitz() {
#if defined(__gfx1250__) && __has_builtin(__builtin_amdgcn_s_wait_asynccnt)
    __builtin_amdgcn_s_wait_asynccnt(0);
#endif
}

// ---------------- rowsum: S[b,i] = sum_j A[b,i,j] + 1e-8 ----------------
__global__ void rowsum_kernel(const float* __restrict__ A, float* __restrict__ S) {
    int wave = (blockIdx.x * blockDim.x + threadIdx.x) >> 5;
    int lane = threadIdx.x & 31;
    const float* row = A + (size_t)wave * Cdim;
    float s = 0.f;
    for (int j = lane; j < Cdim; j += 32) s += row[j];
    #pragma unroll
    for (int m = 16; m; m >>= 1) s += __shfl_xor(s, m, 32);
    if (lane == 0) S[wave] = s + 1e-8f;
}

// ------- GEMM1: C = A(MxK) @ B(KxN) (+bias), LDS double-buffered --------
// Workgroup: 256 thr = 8 waves (4x2), tile 64x128, K-chunk 32.
#define KC 32
__global__ void gemm_wmma_kernel(const float* __restrict__ A,
                                 const float* __restrict__ Bm,
                                 const float* __restrict__ bias,
                                 float* __restrict__ Cout,
                                 int M, int N, int K) {
    __shared__ float lA[2][64][36];    // stride 36: conflict-free A frags, 16B rows
    __shared__ float lB[2][KC][136];   // stride 136: halves hit disjoint bank ranges
    int tid  = threadIdx.x;
    int wave = tid >> 5, lane = tid & 31;
    int wm = wave >> 1, wn = wave & 1;
    int ntiles = N >> 7;
    int m0 = ((int)blockIdx.x / ntiles) * 64;
    int n0 = ((int)blockIdx.x % ntiles) * 128;
    int l15 = lane & 15, half = lane >> 4;

    // stage chunk 0
    {
        const float* Ag = A + (size_t)m0 * K;
        #pragma unroll
        for (int p = 0; p < 2; ++p) {
            int idx = tid + 256 * p;               // 512 float4s
            int r = idx >> 3, c4 = idx & 7;
            g2l_b128(Ag + (size_t)r * K + c4 * 4, &lA[0][r][c4 * 4]);
        }
        const float* Bg = Bm + n0;
        #pragma unroll
        for (int p = 0; p < 4; ++p) {
            int idx = tid + 256 * p;               // 1024 float4s
            int r = idx >> 5, c4 = idx & 31;
            g2l_b128(Bg + (size_t)r * N + c4 * 4, &lB[0][r][c4 * 4]);
        }
    }
    async_waitz();
    __syncthreads();

    v8f acc[4] = {};
    int nchunks = K / KC;
    for (int ch = 0; ch < nchunks; ++ch) {
        int cur = ch & 1;
        if (ch + 1 < nchunks) {                    // prefetch next chunk (async)
            int kc = (ch + 1) * KC;
            const float* Ag = A + (size_t)m0 * K + kc;
            #pragma unroll
            for (int p = 0; p < 2; ++p) {
                int idx = tid + 256 * p;
                int r = idx >> 3, c4 = idx & 7;
                g2l_b128(Ag + (size_t)r * K + c4 * 4, &lA[cur ^ 1][r][c4 * 4]);
            }
            const float* Bg = Bm + (size_t)kc * N + n0;
            #pragma unroll
            for (int p = 0; p < 4; ++p) {
                int idx = tid + 256 * p;
                int r = idx >> 5, c4 = idx & 31;
                g2l_b128(Bg + (size_t)r * N + c4 * 4, &lB[cur ^ 1][r][c4 * 4]);
            }
        }
        const float* la = &lA[cur][wm * 16 + l15][half * 2];
        #pragma unroll
        for (int kk = 0; kk < KC; kk += 4) {
            v2f af; af.x = la[kk]; af.y = la[kk + 1];
            #pragma unroll
            for (int nb = 0; nb < 4; ++nb) {
                int n = wn * 64 + nb * 16 + l15;
                v2f bf;
                bf.x = lB[cur][kk + half * 2][n];
                bf.y = lB[cur][kk + half * 2 + 1][n];
                acc[nb] = __builtin_amdgcn_wmma_f32_16x16x4_f32(
                    false, af, false, bf, (short)0, acc[nb], false, false);
            }
        }
        async_waitz();
        __syncthreads();
    }

    int row0 = m0 + wm * 16 + half * 8;            // C layout: vgpr v -> M=v+8*half
    #pragma unroll
    for (int nb = 0; nb < 4; ++nb) {
        int col = n0 + wn * 64 + nb * 16 + l15;
        float bv = bias ? bias[col] : 0.f;
        #pragma unroll
        for (int v = 0; v < 8; ++v)
            Cout[(size_t)(row0 + v) * N + col] = acc[nb][v] + bv;
    }
}

// ---------------- e_src/e_dst: dot(Wh[b,c,h,:], a_src/a_dst[h]) ---------
__global__ void edot_kernel(const float* __restrict__ Wh,
                            const float* __restrict__ avec,
                            float* __restrict__ esrc, float* __restrict__ edst) {
    int wave = (blockIdx.x * blockDim.x + threadIdx.x) >> 5;   // ((b*C+c)*NH+h)
    int lane = threadIdx.x & 31;
    int h = wave & (NH - 1);
    const float* wrow = Wh + (size_t)wave * OF;
    const float* asrc = avec + h * (2 * OF);
    const float* adst = asrc + OF;
    float s = 0.f, d = 0.f;
    for (int f = lane; f < OF; f += 32) {
        float w = wrow[f];
        s += w * asrc[f];
        d += w * adst[f];
    }
    #pragma unroll
    for (int m = 16; m; m >>= 1) {
        s += __shfl_xor(s, m, 32);
        d += __shfl_xor(d, m, 32);
    }
    if (lane == 0) { esrc[wave] = s; edst[wave] = d; }
}

// ---------------- attention row softmax -> alpha ------------------------
#define ZPAD 1032
__global__ void attn_kernel(const float* __restrict__ A,
                            const float* __restrict__ esrc,
                            const float* __restrict__ edst,
                            const float* __restrict__ S,
                            float* __restrict__ alpha) {
    __shared__ float z[NH * ZPAD];
    __shared__ float invs[NH];
    int bi = blockIdx.x;
    int b  = bi >> 10;
    int i  = bi & 1023;
    int h    = threadIdx.x >> 5;
    int lane = threadIdx.x & 31;

    const float* Arow = A + (size_t)bi * Cdim;
    float invS = 1.0f / S[bi];
    float es   = esrc[bi * NH + h];
    float* zh  = z + h * ZPAD;

    float mx = -3.4e38f;
    for (int j = lane; j < Cdim; j += 32) {
        float aval = Arow[j];
        float e = es + edst[((size_t)b * Cdim + j) * NH + h];
        e = e > 0.f ? e : 0.2f * e;
        bool keep = (aval > 0.f) || (j == i);
        e = keep ? e : -1.0e9f;
        float zz = (e + aval * invS) * TEMP_INV;
        zh[j] = zz;
        mx = fmaxf(mx, zz);
    }
    #pragma unroll
    for (int m = 16; m; m >>= 1) mx = fmaxf(mx, __shfl_xor(mx, m, 32));

    float sum = 0.f;
    for (int j = lane; j < Cdim; j += 32) {
        float ez = __expf(zh[j] - mx);
        zh[j] = ez;
        sum += ez;
    }
    #pragma unroll
    for (int m = 16; m; m >>= 1) sum += __shfl_xor(sum, m, 32);
    if (lane == 0) invs[h] = 1.0f / sum;
    __syncthreads();

    float* arow = alpha + (size_t)bi * (Cdim * NH);
    for (int p = threadIdx.x; p < Cdim * NH; p += 256) {
        int hh = p & (NH - 1);
        int j  = p >> 3;
        arow[p] = z[hh * ZPAD + j] * invs[hh];
    }
}

// ------- GEMM2: X[b,i,h*128+f] += sum_j alpha[b,i,j,h]*Wh[b,j,h,f] ------
// Workgroup: 256 thr, batch b, 16-row i-tile; wave w = head w (16x128 out).
// Alpha tile (16 rows x 256 contiguous floats = all 8 heads, 32 j) staged in
// LDS: each alpha byte read from HBM exactly once, fully coalesced.
__global__ void gemm2_kernel(const float* __restrict__ alpha,
                             const float* __restrict__ Wh,
                             float* __restrict__ X) {
    __shared__ float lA[2][16][260];   // 260: 16B-aligned rows, 4*(m) bank spread
    int tid  = threadIdx.x;
    int h    = tid >> 5;               // wave index == head
    int lane = tid & 31;
    int b  = (int)blockIdx.x >> 6;
    int i0 = ((int)blockIdx.x & 63) * 16;
    int l15 = lane & 15, half = lane >> 4;

    const float* Ab  = alpha + (size_t)b * (Cdim * Cdim * NH)
                             + (size_t)i0 * (Cdim * NH);
    const float* Whb = Wh + (size_t)b * (Cdim * NF) + h * OF;

    #pragma unroll
    for (int p = 0; p < 4; ++p) {                  // 1024 float4s: 16 x 256 floats
        int idx = tid + 256 * p;
        int r = idx >> 6, c4 = idx & 63;
        g2l_b128(Ab + (size_t)r * (Cdim * NH) + c4 * 4, &lA[0][r][c4 * 4]);
    }
    async_waitz();
    __syncthreads();

    v8f acc[8] = {};
    const int nch = Cdim / 32;                     // 32 chunks of 32 j-values
    for (int ch = 0; ch < nch; ++ch) {
        int cur = ch & 1;
        if (ch + 1 < nch) {                        // async prefetch next alpha tile
            const float* Ag = Ab + (size_t)(ch + 1) * (32 * NH);
            #pragma unroll
            for (int p = 0; p < 4; ++p) {
                int idx = tid + 256 * p;
                int r = idx >> 6, c4 = idx & 63;
                g2l_b128(Ag + (size_t)r * (Cdim * NH) + c4 * 4,
                         &lA[cur ^ 1][r][c4 * 4]);
            }
        }
        int jbase = ch * 32;
        __builtin_prefetch(Whb + (size_t)(jbase + 32) * NF, 0, 1);
        const float* la = &lA[cur][l15][h];        // col = jj*8 + h
        #pragma unroll
        for (int kk = 0; kk < 32; kk += 4) {
            int kg = kk + half * 2;
            v2f af;
            af.x = la[kg * 8];
            af.y = la[kg * 8 + 8];
            const float* bp = Whb + (size_t)(jbase + kg) * NF + l15;
            v2f bf[8];
            #pragma unroll
            for (int nb = 0; nb < 8; ++nb) {       // batch B loads: 1 wait / 8 wmma
                bf[nb].x = bp[nb * 16];
                bf[nb].y = bp[nb * 16 + NF];
            }
            #pragma unroll
            for (int nb = 0; nb < 8; ++nb)
                acc[nb] = __builtin_amdgcn_wmma_f32_16x16x4_f32(
                    false, af, false, bf[nb], (short)0, acc[nb], false, false);
        }
        async_waitz();
        __syncthreads();
    }

    int row0 = i0 + half * 8;
    float* xb = X + (size_t)b * (Cdim * NF) + h * OF;
    #pragma unroll
    for (int nb = 0; nb < 8; ++nb) {
        int col = nb * 16 + l15;
        #pragma unroll
        for (int v = 0; v < 8; ++v) {
            size_t idx = (size_t)(row0 + v) * NF + col;
            xb[idx] += acc[nb][v];                 // += residual (R preloaded)
        }
    }
}

// ---------------- LayerNorm + ReLU over NF=1024 -------------------------
__global__ void ln_kernel(const float* __restrict__ X,
                          const float* __restrict__ gamma,
                          const float* __restrict__ beta,
                          float* __restrict__ out) {
    __shared__ float rs[8], rss[8], stats[2];
    int bi = blockIdx.x;
    int h    = threadIdx.x >> 5;
    int lane = threadIdx.x & 31;
    const float* x = X + (size_t)bi * NF;

    float vals[4];
    float s = 0.f, ss = 0.f;
    #pragma unroll
    for (int q = 0; q < 4; ++q) {
        float v = x[threadIdx.x + 256 * q];
        vals[q] = v;
        s += v; ss += v * v;
    }
    #pragma unroll
    for (int m = 16; m; m >>= 1) {
        s  += __shfl_xor(s,  m, 32);
        ss += __shfl_xor(ss, m, 32);
    }
    if (lane == 0) { rs[h] = s; rss[h] = ss; }
    __syncthreads();
    if (threadIdx.x == 0) {
        float ts = 0.f, tss = 0.f;
        #pragma unroll
        for (int w = 0; w < 8; ++w) { ts += rs[w]; tss += rss[w]; }
        float mu  = ts * (1.0f / NF);
        float var = tss * (1.0f / NF) - mu * mu;
        stats[0] = mu;
        stats[1] = __frsqrt_rn(var + LN_EPS);
    }
    __syncthreads();
    float mu = stats[0], rstd = stats[1];
    float* o = out + (size_t)bi * NF;
    #pragma unroll
    for (int q = 0; q < 4; ++q) {
        int f = threadIdx.x + 256 * q;
        float v = (vals[q] - mu) * rstd * gamma[f] + beta[f];
        o[f] = fmaxf(v, 0.f);
    }
}

// ---------------- launch ------------------------------------------------
extern "C" void kernel_launch(void* const* d_in, const int* in_sizes, int n_in,
                              void* d_out, int out_size, void* d_ws, size_t ws_size,
                              hipStream_t stream) {
    const float* H     = (const float*)d_in[0];
    const float* A     = (const float*)d_in[1];
    const float* W     = (const float*)d_in[2];
    const float* avec  = (const float*)d_in[3];
    const float* resW  = (const float*)d_in[4];
    const float* resB  = (const float*)d_in[5];
    const float* gamma = (const float*)d_in[6];
    const float* beta  = (const float*)d_in[7];

    float* out0  = (float*)d_out;
    float* alpha = (float*)d_out + (size_t)Bsz * Cdim * NF;

    float* ws   = (float*)d_ws;
    float* Wh   = ws;
    float* Rx   = ws + (size_t)8388608;
    float* esrc = ws + (size_t)16777216;
    float* edst = ws + (size_t)16842752;
    float* S    = ws + (size_t)16908288;

    const int M = Bsz * Cdim;

    rowsum_kernel<<<1024, 256, 0, stream>>>(A, S);
    gemm_wmma_kernel<<<1024, 256, 0, stream>>>(H, W, nullptr, Wh, M, NF, INF);
    gemm_wmma_kernel<<<1024, 256, 0, stream>>>(H, resW, resB, Rx, M, NF, INF);
    edot_kernel<<<8192, 256, 0, stream>>>(Wh, avec, esrc, edst);
    attn_kernel<<<8192, 256, 0, stream>>>(A, esrc, edst, S, alpha);
    gemm2_kernel<<<512, 256, 0, stream>>>(alpha, Wh, Rx);
    ln_kernel<<<8192, 256, 0, stream>>>(Rx, gamma, beta, out0);
}